// AIGTripleEmbeddingLayer_67482526154762
// MI455X (gfx1250) — compile-verified
//
#include <hip/hip_runtime.h>

// ---------------------------------------------------------------------------
// AIG triple-embedding layer, refactored for MI455X (gfx1250):
//   out[t] = Psrc[flat(src)] + Rtab[rel] + Pdst[flat(dst)]
// where Psrc = table @ Ws^T, Pdst = table @ Wd^T, Rtab = edge_emb @ Wr^T + b.
// Only referenced table rows are projected (bitmap of used gate rows), so the
// WMMA work collapses to ~0.1 GFLOP and the kernel runs at the 256 MB output-
// write roofline (~11 us at 23.3 TB/s).
// ---------------------------------------------------------------------------

typedef __attribute__((ext_vector_type(2))) float v2f;
typedef __attribute__((ext_vector_type(4))) float v4f;
typedef __attribute__((ext_vector_type(8))) float v8f;

#define DD 128            // embedding dim
#define WSTRIDE 384       // W row stride (3*D)
#define LDS_STRIDE 132    // 128 + 4 pad: row->row bank offset = 4 (64 banks)
#define NICAP 1024        // host-side cap for num_input_nodes (actual = 256)
#define NOCAP 1024        // host-side cap for num_output_nodes (actual = 256)

__device__ __forceinline__ int flat_index(int ty, int ix, int ni, int no, int ngCap) {
    // table layout: [input(ni) | output(no) | gate(ngCap) | zero(1)]
    if (ty == 0) { int i = ix < 0 ? 0 : (ix >= ni ? ni - 1 : ix); return i; }
    if (ty == 1) { int i = ix < 0 ? 0 : (ix >= no ? no - 1 : ix); return ni + i; }
    if (ty == 2) { int i = ix < 0 ? 0 : (ix >= ngCap ? ngCap - 1 : ix); return ni + no + i; }
    return ni + no + ngCap; // type 3 -> zero row
}

// ---- Pass A: mark used gate rows ------------------------------------------
__global__ __launch_bounds__(256)
void mark_kernel(const int* __restrict__ st, const int* __restrict__ si,
                 const int* __restrict__ dt, const int* __restrict__ di,
                 unsigned* __restrict__ bitmap, int T, int ngCap) {
    int t = blockIdx.x * 256 + threadIdx.x;
    if (t >= T) return;
    if (st[t] == 2) {
        int g = si[t]; g = g < 0 ? 0 : (g >= ngCap ? ngCap - 1 : g);
        atomicOr(&bitmap[g >> 5], 1u << (g & 31));
    }
    if (dt[t] == 2) {
        int g = di[t]; g = g < 0 ? 0 : (g >= ngCap ? ngCap - 1 : g);
        atomicOr(&bitmap[g >> 5], 1u << (g & 31));
    }
}

// ---- Rtab: edge_emb @ Wr^T + b  (2x128, trivial) --------------------------
__global__ __launch_bounds__(256)
void rtab_kernel(const float* __restrict__ edge_emb, const float* __restrict__ W,
                 const float* __restrict__ bias, float* __restrict__ rtab) {
    int tid = threadIdx.x;           // 256 threads = 2 rows x 128 cols
    int r = tid >> 7;
    int i = tid & 127;
    float acc = bias[i];
    const float* wr = W + (size_t)i * WSTRIDE + DD;  // Wr block
    const float* ev = edge_emb + r * DD;
#pragma unroll 8
    for (int k = 0; k < DD; ++k) acc += ev[k] * wr[k];
    rtab[r * DD + i] = acc;
}

// ---- Pass B: project used table rows with fp32 WMMA -----------------------
// 256 threads = 8 waves; wave w owns output columns [16w, 16w+16).
// A operand (16x4 fp32): lane L -> M = L%16, VGPR j -> K = j + 2*(L>=16).
// B operand (4x16 fp32): lane L -> N = L%16, VGPR j -> K = j + 2*(L>=16).
// C/D (16x16 fp32, ISA 7.12.2): lane L -> N = L%16, VGPR j -> M = j + 8*(L>=16).
__global__ __launch_bounds__(256)
void project_kernel(const float* __restrict__ gate_emb, const float* __restrict__ W,
                    const unsigned* __restrict__ bitmap,
                    const int* __restrict__ pni, const int* __restrict__ pno,
                    int ngCap,
                    float* __restrict__ psrc, float* __restrict__ pdst) {
    __shared__ float lds[16 * LDS_STRIDE];
    __shared__ int s_flag;

    int tid = threadIdx.x;
    int ni = *pni; ni = ni < 1 ? 1 : (ni > NICAP ? NICAP : ni);
    int no = *pno; no = no < 1 ? 1 : (no > NOCAP ? NOCAP : no);
    int nio = ni + no;
    int NT = nio + ngCap + 1;
    int row0 = blockIdx.x * 16;
    if (row0 >= NT) return;

    if (tid == 0) s_flag = 0;
    __syncthreads();
    if (tid < 16) {
        int r = row0 + tid;
        bool need = false;
        if (r < NT) {
            if (r < nio || r == NT - 1) need = true;
            else {
                int g = r - nio;
                need = (bitmap[g >> 5] >> (g & 31)) & 1u;
            }
        }
        if (need) s_flag = 1;
    }
    __syncthreads();
    if (!s_flag) return;   // tile holds no referenced rows -> skip

    // Build 16x128 A tile in LDS (sinusoid rows synthesized, gate rows loaded)
    for (int e = tid; e < 16 * DD; e += 256) {
        int row = e >> 7;
        int col = e & 127;
        int r = row0 + row;
        float v = 0.0f;
        if (r < NT - 1) {
            if (r < nio) {
                int p = (r < ni) ? r : (r - ni);
                int j = col >> 1;
                // 10000^(-2j/128): exp(-(2j)*ln(10000)/128)
                float ang = (float)p * __expf((float)(2 * j) * (-9.210340371976184f / 128.0f));
                v = (col & 1) ? __cosf(ang) : __sinf(ang);
            } else {
                v = gate_emb[(size_t)(r - nio) * DD + col];
            }
        }
        lds[row * LDS_STRIDE + col] = v;
    }
    __syncthreads();

    int lane = tid & 31;
    int w = tid >> 5;
    int m = lane & 15;
    int hi = lane >> 4;          // 0 or 1
    int kofs = hi * 2;
    int iglob = w * 16 + m;      // output column N handled by this lane

    const float* wS = W + (size_t)iglob * WSTRIDE + kofs;        // Ws block
    const float* wDp = wS + 2 * DD;                              // Wd block (+256)
    const float* ap = lds + m * LDS_STRIDE + kofs;

    v8f accS = {};
    v8f accD = {};
#pragma unroll 4
    for (int s = 0; s < 32; ++s) {
        v2f a;  a.x = ap[s * 4 + 0];  a.y = ap[s * 4 + 1];
        v2f bs; bs.x = wS[s * 4 + 0]; bs.y = wS[s * 4 + 1];
        v2f bd; bd.x = wDp[s * 4 + 0]; bd.y = wDp[s * 4 + 1];
        accS = __builtin_amdgcn_wmma_f32_16x16x4_f32(false, a, false, bs,
                                                     (short)0, accS, false, false);
        accD = __builtin_amdgcn_wmma_f32_16x16x4_f32(false, a, false, bd,
                                                     (short)0, accD, false, false);
    }

#pragma unroll
    for (int j = 0; j < 8; ++j) {
        int rout = row0 + j + hi * 8;
        if (rout < NT) {
            size_t o = (size_t)rout * DD + iglob;
            psrc[o] = accS[j];
            pdst[o] = accD[j];
        }
    }
}

// ---- Pass C: gather + add, streaming non-temporal output ------------------
__global__ __launch_bounds__(256)
void gather_kernel(const float* __restrict__ psrc, const float* __restrict__ pdst,
                   const float* __restrict__ rtab,
                   const int* __restrict__ si, const int* __restrict__ st,
                   const int* __restrict__ rel,
                   const int* __restrict__ di, const int* __restrict__ dt,
                   const int* __restrict__ pni, const int* __restrict__ pno,
                   int ngCap, int T, float* __restrict__ out) {
    int e = blockIdx.x * 8 + (threadIdx.x >> 5);   // one wave32 per edge row
    if (e >= T) return;
    int lane = threadIdx.x & 31;                   // 32 lanes x float4 = 128

    int ni = *pni; ni = ni < 1 ? 1 : (ni > NICAP ? NICAP : ni);
    int no = *pno; no = no < 1 ? 1 : (no > NOCAP ? NOCAP : no);

    int fs = flat_index(st[e], si[e], ni, no, ngCap);
    int fd = flat_index(dt[e], di[e], ni, no, ngCap);
    int rr = rel[e] & 1;

    const v4f* ps = (const v4f*)(psrc + (size_t)fs * DD) + lane;  // L2-hot
    const v4f* pd = (const v4f*)(pdst + (size_t)fd * DD) + lane;  // L2-hot
    const v4f* pr = (const v4f*)(rtab + rr * DD) + lane;          // 1KB, hot

    v4f a = *ps, b = *pd, c = *pr;
    v4f o = a + b + c;
    // 256 MB streaming store: non-temporal so it doesn't evict the tables
    __builtin_nontemporal_store(o, (v4f*)(out + (size_t)e * DD) + lane);
}

// ---------------------------------------------------------------------------
extern "C" void kernel_launch(void* const* d_in, const int* in_sizes, int n_in,
                              void* d_out, int out_size, void* d_ws, size_t ws_size,
                              hipStream_t stream) {
    const float* gate_emb = (const float*)d_in[0];
    const float* edge_emb = (const float*)d_in[1];
    const float* W        = (const float*)d_in[2];
    const float* bias     = (const float*)d_in[3];
    const int*   src_idx  = (const int*)d_in[4];
    const int*   src_type = (const int*)d_in[5];
    const int*   rel      = (const int*)d_in[6];
    const int*   dst_idx  = (const int*)d_in[7];
    const int*   dst_type = (const int*)d_in[8];
    const int*   pni      = (const int*)d_in[9];   // device scalar
    const int*   pno      = (const int*)d_in[10];  // device scalar

    const int T  = in_sizes[4];
    const int NG = in_sizes[0] / DD;

    // Workspace layout: [bitmap | rtab | Psrc | Pdst]
    char* ws = (char*)d_ws;
    size_t bmWords = (size_t)(NG + 31) / 32;
    size_t bmBytes = (bmWords * 4 + 255) & ~(size_t)255;
    size_t rtBytes = (2 * DD * 4 + 255) & ~(size_t)255;
    size_t fixed   = bmBytes + rtBytes;

    int ngCap = NG;
    size_t rowsCap = (size_t)NICAP + NOCAP + (size_t)NG + 1;
    if (ws_size < fixed + 2 * rowsCap * DD * 4) {
        // Shrink gate capacity to fit (clamped gate indices remain consistent
        // between mark/project/gather, so behavior stays deterministic).
        size_t avail = (ws_size > fixed) ? (ws_size - fixed) : 0;
        long rows = (long)(avail / (2 * (size_t)DD * 4));
        long ngc  = rows - (NICAP + NOCAP + 1);
        if (ngc < 1) ngc = 1;
        if (ngc > NG) ngc = NG;
        ngCap = (int)ngc;
        rowsCap = (size_t)NICAP + NOCAP + (size_t)ngCap + 1;
    }

    unsigned* bitmap = (unsigned*)ws;
    float* rtab = (float*)(ws + bmBytes);
    float* psrc = (float*)(ws + fixed);
    float* pdst = psrc + rowsCap * DD;

    if (T <= 0) return;

    hipMemsetAsync(bitmap, 0, bmWords * 4, stream);

    mark_kernel<<<(T + 255) / 256, 256, 0, stream>>>(
        src_type, src_idx, dst_type, dst_idx, bitmap, T, ngCap);

    rtab_kernel<<<1, 256, 0, stream>>>(edge_emb, W, bias, rtab);

    int tiles = (int)(((size_t)NICAP + NOCAP + (size_t)ngCap + 1 + 15) / 16);
    project_kernel<<<tiles, 256, 0, stream>>>(
        gate_emb, W, bitmap, pni, pno, ngCap, psrc, pdst);

    gather_kernel<<<(T + 7) / 8, 256, 0, stream>>>(
        psrc, pdst, rtab, src_idx, src_type, rel, dst_idx, dst_type,
        pni, pno, ngCap, T, d_out ? (float*)d_out : psrc);
}